// DifferentiableChebyshevOperator_81166291960318
// MI455X (gfx1250) — compile-verified
//
#include <hip/hip_runtime.h>
#include <math.h>

#ifndef M_PI
#define M_PI 3.14159265358979323846
#endif
#ifndef __has_builtin
#define __has_builtin(x) 0
#endif

#define EPSF     0.01f
#define D_FEAT   128
#define CHUNK    2048          // edges staged per block
#define WAVES    8
#define PER_WAVE (CHUNK / WAVES)  // 256 edges per wave
#define BLOCK    256
#define M_CHEB   30

typedef unsigned int v4u __attribute__((ext_vector_type(4)));
typedef int          v8i __attribute__((ext_vector_type(8)));
typedef int          v4i __attribute__((ext_vector_type(4)));

#if __has_builtin(__builtin_amdgcn_tensor_load_to_lds)
#define USE_TDM 1
#else
#define USE_TDM 0
#endif

// ---- diagnostic probe: FIRST in TU so the disasm snippet reveals how
// atomicAdd(float*) lowers on gfx1250 (want: global_atomic_add_f32, no CAS loop).
// Never launched; costs nothing.
__global__ void atom_probe(float* __restrict__ p, const float* __restrict__ v) {
  atomicAdd(p + threadIdx.x, v[threadIdx.x]);
}

__device__ __forceinline__ unsigned rfl(unsigned x) {
  return (unsigned)__builtin_amdgcn_readfirstlane((int)x);
}

#if USE_TDM
// 1-D TDM load: n_elems 4-byte elements from gptr -> LDS byte offset lds_off.
// Descriptor per CDNA5 ISA ch.7/8 (D# group0 128b, group1 256b; groups 2/3 zero).
__device__ __forceinline__ void tdm_load_1d(unsigned lds_off, const void* gptr,
                                            unsigned n_elems) {
  unsigned long long ga = (unsigned long long)(uintptr_t)gptr;
  v4u g0;
  g0[0] = rfl(1u);                               // count=1, is_restore=0, gather=0
  g0[1] = rfl(lds_off);                          // lds_addr (bytes)
  g0[2] = rfl((unsigned)ga);                     // global_addr[31:0]
  g0[3] = rfl(((unsigned)(ga >> 32) & 0x1FFFFFFu) | (2u << 30)); // addr[56:32] | type=2
  v8i g1;
  g1[0] = (int)rfl(2u << 16);                    // data_size = 4B (code 2)
  g1[1] = (int)rfl((n_elems & 0xFFFFu) << 16);   // tensor_dim0[15:0]
  g1[2] = (int)rfl(((n_elems >> 16) & 0xFFFFu) | (1u << 16)); // dim0 hi | tensor_dim1=1
  g1[3] = (int)rfl((n_elems & 0xFFFFu) << 16);   // tile_dim0 = n_elems
  g1[4] = 0;                                     // tile_dim1=0 (unused), tile_dim2=0
  g1[5] = (int)rfl(n_elems);                     // tensor_dim0_stride lo
  g1[6] = 0;
  g1[7] = 0;
  v4i gz = {0, 0, 0, 0};
#if defined(__clang_major__) && (__clang_major__ >= 23)
  v8i gz8 = {0, 0, 0, 0, 0, 0, 0, 0};
  __builtin_amdgcn_tensor_load_to_lds(g0, g1, gz, gz, gz8, 0);
#else
  __builtin_amdgcn_tensor_load_to_lds(g0, g1, gz, gz, 0);
#endif
}
#endif

// ---------------- SpMM scatter: Out += wn[e] * Y[cols[e]] at row rows[e] ----
// One wave per edge iteration: 32 lanes x float4 = one 128-float row.
// Edge arrays are DMA-staged into LDS with the Tensor Data Mover per wave.

__global__ __launch_bounds__(BLOCK) void spmm_kernel(
    const int* __restrict__ rows, const int* __restrict__ cols,
    const float* __restrict__ wn, const float* __restrict__ Y,
    float* __restrict__ Out, int E) {
  __shared__ int   sRow[CHUNK];
  __shared__ int   sCol[CHUNK];
  __shared__ float sW[CHUNK];

  const int wave = threadIdx.x >> 5;
  const int lane = threadIdx.x & 31;
  const int base = blockIdx.x * CHUNK + wave * PER_WAVE;
  if (base >= E) return;                  // wave-uniform; no barriers used below
  const int n = min(PER_WAVE, E - base);
  const int sbase = wave * PER_WAVE;

#if USE_TDM
  tdm_load_1d((unsigned)(uintptr_t)&sCol[sbase], cols + base, (unsigned)n);
  tdm_load_1d((unsigned)(uintptr_t)&sRow[sbase], rows + base, (unsigned)n);
  tdm_load_1d((unsigned)(uintptr_t)&sW[sbase],   wn   + base, (unsigned)n);
  __builtin_amdgcn_s_wait_tensorcnt(0);
  asm volatile("" ::: "memory");          // TDM wrote LDS behind LLVM's back
#else
  for (int i = lane; i < n; i += 32) {
    sCol[sbase + i] = cols[base + i];
    sRow[sbase + i] = rows[base + i];
    sW[sbase + i]   = wn[base + i];
  }
  __threadfence_block();
#endif

  for (int i = 0; i < n; ++i) {
    const int   c = sCol[sbase + i];
    const int   r = sRow[sbase + i];
    const float w = sW[sbase + i];
    if (i + 8 < n) {  // pre-touch gather row 8 edges ahead (global_prefetch_b8)
      __builtin_prefetch(Y + (size_t)sCol[sbase + i + 8] * D_FEAT + lane * 4, 0, 3);
    }
    const float4 y = *(const float4*)(Y + (size_t)c * D_FEAT + lane * 4);
    float* o = Out + (size_t)r * D_FEAT + lane * 4;
    atomicAdd(o + 0, w * y.x);
    atomicAdd(o + 1, w * y.y);
    atomicAdd(o + 2, w * y.z);
    atomicAdd(o + 3, w * y.w);
  }
}

// ---------------- precompute kernels ----------------

__global__ __launch_bounds__(BLOCK) void edge_prep1(
    const int* __restrict__ rows, const float* __restrict__ Wv,
    const float* __restrict__ kap, const float* __restrict__ alpha,
    const float* __restrict__ center, float* __restrict__ wn,
    float* __restrict__ deg, int E) {
  int e = blockIdx.x * blockDim.x + threadIdx.x;
  if (e >= E) return;
  float a = alpha[0];
  float sens = fmaxf(a, 0.0f) + log1pf(expf(-fabsf(a)));   // stable softplus
  float z = sens * (kap[e] - center[0]);
  float sig = 1.0f / (1.0f + expf(-z));
  float wp = Wv[e] * (EPSF + (1.0f - EPSF) * sig);
  wn[e] = wp;
  atomicAdd(&deg[rows[e]], wp);
}

__global__ __launch_bounds__(BLOCK) void node_dinv(float* __restrict__ deg, int N) {
  int i = blockIdx.x * blockDim.x + threadIdx.x;
  if (i >= N) return;
  deg[i] = 1.0f / (sqrtf(deg[i]) + 1e-8f);
}

__global__ __launch_bounds__(BLOCK) void edge_prep2(
    const int* __restrict__ rows, const int* __restrict__ cols,
    const float* __restrict__ dis, float* __restrict__ wn, int E) {
  int e = blockIdx.x * blockDim.x + threadIdx.x;
  if (e >= E) return;
  // fold L_rescaled = -W_norm sign into the edge weight
  wn[e] = -(dis[rows[e]] * wn[e] * dis[cols[e]]);
}

// ---------------- elementwise combines ----------------

__global__ __launch_bounds__(BLOCK) void init_acc(
    const float4* __restrict__ X, const float4* __restrict__ T1,
    float4* __restrict__ acc, float c0, float c1, int n4) {
  int i = blockIdx.x * blockDim.x + threadIdx.x;
  if (i >= n4) return;
  float4 x = X[i], t = T1[i], a;
  a.x = c0 * x.x + c1 * t.x;
  a.y = c0 * x.y + c1 * t.y;
  a.z = c0 * x.z + c1 * t.z;
  a.w = c0 * x.w + c1 * t.w;
  acc[i] = a;
}

// S holds L@T_cur (sign already folded). T_k = 2*S - T_prev (in place); acc += ck*T_k.
__global__ __launch_bounds__(BLOCK) void combine(
    float4* __restrict__ S, const float4* __restrict__ Tprev,
    float4* __restrict__ acc, float ck, int n4) {
  int i = blockIdx.x * blockDim.x + threadIdx.x;
  if (i >= n4) return;
  float4 s = S[i], p = Tprev[i], t, a = acc[i];
  t.x = 2.0f * s.x - p.x;
  t.y = 2.0f * s.y - p.y;
  t.z = 2.0f * s.z - p.z;
  t.w = 2.0f * s.w - p.w;
  S[i] = t;
  a.x += ck * t.x; a.y += ck * t.y; a.z += ck * t.z; a.w += ck * t.w;
  acc[i] = a;
}

// ---------------- host driver ----------------

extern "C" void kernel_launch(void* const* d_in, const int* in_sizes, int n_in,
                              void* d_out, int out_size, void* d_ws, size_t ws_size,
                              hipStream_t stream) {
  const int*   Widx   = (const int*)d_in[0];
  const float* Wv     = (const float*)d_in[1];
  const float* kap    = (const float*)d_in[2];
  const float* X      = (const float*)d_in[3];
  const float* alpha  = (const float*)d_in[4];
  const float* center = (const float*)d_in[5];

  const int E  = in_sizes[1];
  const int ND = in_sizes[3];
  const int N  = ND / D_FEAT;
  const int* rows = Widx;
  const int* cols = Widx + E;
  float* acc = (float*)d_out;

  // workspace carve-out
  char* ws = (char*)d_ws;
  size_t off = 0;
  auto carve = [&](size_t bytes) {
    void* p = ws + off;
    off += (bytes + 255) & ~(size_t)255;
    return p;
  };
  float* wn  = (float*)carve((size_t)E * 4);
  float* deg = (float*)carve((size_t)N * 4);
  float* B1  = (float*)carve((size_t)ND * 4);
  float* B2  = (float*)carve((size_t)ND * 4);
  float* B3  = (float*)carve((size_t)ND * 4);
  if (off > ws_size) return;  // workspace too small; nothing safe to do

  // Chebyshev-Gauss coefficients for exp(-5*lam), lambda_max=2 (trace-time constants)
  double coeffs[M_CHEB];
  {
    double f[M_CHEB], th[M_CHEB];
    for (int j = 0; j < M_CHEB; ++j) {
      th[j] = M_PI * (j + 0.5) / M_CHEB;
      double lam = 1.0 * (cos(th[j]) + 1.0);  // LAMBDA_MAX/2 * (x+1)
      f[j] = exp(-5.0 * lam);                 // T_SCALE = 5
    }
    for (int k = 0; k < M_CHEB; ++k) {
      double s = 0.0;
      for (int j = 0; j < M_CHEB; ++j) s += f[j] * cos(k * th[j]);
      coeffs[k] = 2.0 / M_CHEB * s;
    }
    coeffs[0] *= 0.5;
  }

  const int eb = (E + BLOCK - 1) / BLOCK;
  const int nb = (N + BLOCK - 1) / BLOCK;
  const int n4 = ND / 4;
  const int cb = (n4 + BLOCK - 1) / BLOCK;
  const int spmm_grid = (E + CHUNK - 1) / CHUNK;

  // edge gating + normalization
  hipMemsetAsync(deg, 0, (size_t)N * 4, stream);
  edge_prep1<<<eb, BLOCK, 0, stream>>>(rows, Wv, kap, alpha, center, wn, deg, E);
  node_dinv<<<nb, BLOCK, 0, stream>>>(deg, N);
  edge_prep2<<<eb, BLOCK, 0, stream>>>(rows, cols, deg, wn, E);

  // T1 = L @ X
  hipMemsetAsync(B1, 0, (size_t)ND * 4, stream);
  spmm_kernel<<<spmm_grid, BLOCK, 0, stream>>>(rows, cols, wn, X, B1, E);
  init_acc<<<cb, BLOCK, 0, stream>>>((const float4*)X, (const float4*)B1,
                                     (float4*)acc, (float)coeffs[0],
                                     (float)coeffs[1], n4);

  // Chebyshev recurrence with 3-buffer rotation
  const float* Tprev = X;
  float* Tcur = B1;
  float* bufs[3] = {B1, B2, B3};
  int nextIdx = 1;
  for (int k = 2; k < M_CHEB; ++k) {
    float* Tnext = bufs[nextIdx];
    hipMemsetAsync(Tnext, 0, (size_t)ND * 4, stream);
    spmm_kernel<<<spmm_grid, BLOCK, 0, stream>>>(rows, cols, wn, Tcur, Tnext, E);
    combine<<<cb, BLOCK, 0, stream>>>((float4*)Tnext, (const float4*)Tprev,
                                      (float4*)acc, (float)coeffs[k], n4);
    Tprev = Tcur;
    Tcur = Tnext;
    nextIdx = (nextIdx + 1) % 3;
  }
}